// WaveConv1d_23545010717198
// MI455X (gfx1250) — compile-verified
//
#include <hip/hip_runtime.h>

typedef __attribute__((ext_vector_type(2))) float v2f;
typedef __attribute__((ext_vector_type(8))) float v8f;

// ---------------- filter taps (compile-time constants) ----------------
__constant__ float c_rec_lo[12] = {
    0.11154074335008017f,  0.4946238903983854f,   0.7511339080215775f,
    0.3152503517092432f,  -0.22626469396516913f, -0.12976686756709563f,
    0.09750160558707936f,  0.02752286553001629f, -0.031582039318031156f,
    0.0005538422009938016f, 0.004777257511010651f, -0.00107730108499558f};
__constant__ float c_rec_hi[12] = {
   -0.00107730108499558f, -0.004777257511010651f, 0.0005538422009938016f,
    0.031582039318031156f, 0.02752286553001629f, -0.09750160558707936f,
   -0.12976686756709563f,  0.22626469396516913f,  0.3152503517092432f,
   -0.7511339080215775f,   0.4946238903983854f,  -0.11154074335008017f};
__constant__ float c_dec_lo[12] = {
   -0.00107730108499558f,  0.004777257511010651f, 0.0005538422009938016f,
   -0.031582039318031156f, 0.02752286553001629f,  0.09750160558707936f,
   -0.12976686756709563f, -0.22626469396516913f,  0.3152503517092432f,
    0.7511339080215775f,   0.4946238903983854f,   0.11154074335008017f};
__constant__ float c_dec_hi[12] = {
   -0.11154074335008017f,  0.4946238903983854f,  -0.7511339080215775f,
    0.3152503517092432f,   0.22626469396516913f, -0.12976686756709563f,
   -0.09750160558707936f,  0.02752286553001629f,  0.031582039318031156f,
    0.0005538422009938016f, -0.004777257511010651f, -0.00107730108499558f};

// ---------------- analysis filter bank: stride-2, symmetric pad ----------------
__global__ void afb_kernel(const float* __restrict__ in, float* __restrict__ lo,
                           float* __restrict__ hi, int Nin, int Nout, int pleft,
                           int total) {
  int t = blockIdx.x * blockDim.x + threadIdx.x;
  if (t >= total) return;
  int n  = t % Nout;
  int bc = t / Nout;
  const float* src = in + (size_t)bc * Nin;
  float accLo = 0.f, accHi = 0.f;
  int j0 = 2 * n - pleft;
#pragma unroll
  for (int tap = 0; tap < 12; ++tap) {
    int j = j0 + tap;
    if (j < 0) j = -j - 1;                 // symmetric reflect (edge included)
    else if (j >= Nin) j = 2 * Nin - 1 - j;
    float v = src[j];
    accLo = fmaf(v, c_rec_lo[tap], accLo);
    accHi = fmaf(v, c_rec_hi[tap], accHi);
  }
  lo[(size_t)bc * Nout + n] = accLo;
  hi[(size_t)bc * Nout + n] = accHi;
}

// ---------------- synthesis filter bank: lhs_dilation=2, pad(1,1) ----------------
__global__ void sfb_kernel(const float* __restrict__ y, const float* __restrict__ hi,
                           float* __restrict__ out, int N, int yStride, int Nout,
                           int total) {
  int t = blockIdx.x * blockDim.x + threadIdx.x;
  if (t >= total) return;
  int m  = t % Nout;
  int bc = t / Nout;
  const float* yr = y  + (size_t)bc * yStride;  // trim implicit: only i < N read
  const float* hr = hi + (size_t)bc * N;
  float acc = 0.f;
#pragma unroll
  for (int tap = 0; tap < 12; ++tap) {
    int j = m + tap;
    if (j & 1) {
      int i = (j - 1) >> 1;                 // provably in [0, N)
      acc = fmaf(yr[i], c_dec_lo[tap], acc);
      acc = fmaf(hr[i], c_dec_hi[tap], acc);
    }
  }
  out[(size_t)bc * Nout + m] = acc;
}

// ---------------- per-position GEMM via V_WMMA_F32_16X16X4_F32 ----------------
// Out[b,o,x] = sum_i A[b,i,x] * W[i,o,x], per x in [0,14). A: 256x64, W: 64x64.
// One wave32 owns a 16x16 output tile; K=64 in 16 steps of 4.
// All strides are compile-time (NP=14), and the lane-half K offset is folded into
// the base pointers so unrolled loads use immediate IOFFSETs (batched by scheduler).
__global__ void einsum_wmma(const float* __restrict__ A, const float* __restrict__ W,
                            float* __restrict__ O) {
  constexpr int NP = 14;
  const int x    = blockIdx.x;             // position 0..13
  const int wave = threadIdx.x >> 5;       // 0..7
  const int lane = threadIdx.x & 31;
  const int tile = blockIdx.y * 8 + wave;  // 0..63
  const int bt   = tile >> 2;              // 16 M-tiles over batch 256
  const int ot   = tile & 3;               // 4 N-tiles over Cout 64
  const int half = lane >> 4;              // 0 or 1
  const int l16  = lane & 15;
  const int m    = bt * 16 + l16;
  const int n    = ot * 16 + l16;

  // A frag element (m, k+half*2 + {0,1}); W frag element (k+half*2 + {0,1}, n)
  const float* aBase = A + ((size_t)m * 64 + half * 2) * NP + x;
  const float* wBase = W + ((size_t)(half * 2) * 64 + n) * NP + x;

  v8f c = {};
#pragma unroll
  for (int k = 0; k < 64; k += 4) {
    v2f a, b;
    a.x = aBase[k * NP];
    a.y = aBase[(k + 1) * NP];
    b.x = wBase[k * 64 * NP];
    b.y = wBase[(k + 1) * 64 * NP];
    c = __builtin_amdgcn_wmma_f32_16x16x4_f32(false, a, false, b, (short)0, c,
                                              false, false);
  }

  float* oBase = O + ((size_t)(bt * 16 + half * 8) * 64 + n) * NP + x;
#pragma unroll
  for (int r = 0; r < 8; ++r) {
    oBase[r * 64 * NP] = c[r];
  }
}

// ---------------- host orchestration ----------------
extern "C" void kernel_launch(void* const* d_in, const int* in_sizes, int n_in,
                              void* d_out, int out_size, void* d_ws, size_t ws_size,
                              hipStream_t stream) {
  (void)in_sizes; (void)n_in; (void)out_size; (void)ws_size;
  const float* x  = (const float*)d_in[0];
  const float* W1 = (const float*)d_in[1];
  const float* W2 = (const float*)d_in[2];
  float* out = (float*)d_out;

  const int BC = 256 * 64;
  static const int Ns[9] = {1024, 517, 264, 137, 74, 42, 26, 18, 14};

  // carve workspace
  char* ws = (char*)d_ws;
  size_t off = 0;
  auto alloc = [&](size_t elems) {
    float* p = (float*)(ws + off);
    off += ((elems * sizeof(float) + 255) / 256) * 256;
    return p;
  };
  float* hi[9];
  for (int k = 1; k <= 8; ++k) hi[k] = alloc((size_t)BC * Ns[k]);
  float* lobuf[2] = {alloc((size_t)BC * 517), alloc((size_t)BC * 517)};
  float* mmLo = alloc((size_t)BC * 14);
  float* mmHi = alloc((size_t)BC * 14);
  float* ybuf[2] = {alloc((size_t)BC * 518), alloc((size_t)BC * 518)};

  // ---- analysis chain ----
  const float* cur = x;
  for (int k = 0; k < 8; ++k) {
    const int Nin = Ns[k], Nout = Ns[k + 1];
    float* lo = lobuf[k & 1];
    const int p  = 2 * Nout + 10 - Nin;
    const int pl = p / 2;
    const int total = BC * Nout;
    afb_kernel<<<(total + 255) / 256, 256, 0, stream>>>(cur, lo, hi[k + 1], Nin,
                                                        Nout, pl, total);
    cur = lo;  // after loop: cur holds the length-14 lowpass band
  }

  // ---- per-position 256x64x64 fp32 GEMMs on the WMMA pipe ----
  einsum_wmma<<<dim3(14, 8), 256, 0, stream>>>(cur, W1, mmLo);
  einsum_wmma<<<dim3(14, 8), 256, 0, stream>>>(hi[8], W2, mmHi);

  // ---- synthesis chain (last level's hi replaced by einsum result) ----
  const float* ycur = mmLo;
  int yLen = 14;
  for (int k = 8; k >= 1; --k) {
    const int N = Ns[k];              // hi length == trimmed y length
    const int Nout = 2 * N - 10;
    const float* hband = (k == 8) ? (const float*)mmHi : (const float*)hi[k];
    float* dst = (k == 1) ? out : ybuf[k & 1];
    const int total = BC * Nout;
    sfb_kernel<<<(total + 255) / 256, 256, 0, stream>>>(ycur, hband, dst, N, yLen,
                                                        Nout, total);
    ycur = dst;
    yLen = Nout;
  }
}